// E2GNN_57904749085232
// MI455X (gfx1250) — compile-verified
//
#include <hip/hip_runtime.h>
#include <hip/hip_bf16.h>

#define H   128
#define H2  64
#define H3  384
#define RDIM 64

typedef float v2f __attribute__((ext_vector_type(2)));
typedef float v8f __attribute__((ext_vector_type(8)));

__device__ __forceinline__ float ssilu_f(float v) {
  // silu(v) * (1/0.6)
  return 1.6666666667f * (v / (1.0f + __expf(-v)));
}

// ---------------------------------------------------------------------------
// WMMA f32 GEMM:  out[M,N] = EPI(A[M,K] @ W[K,N] + bias[N], res[M,N])
//   EPI 0: none   1: ssilu   2: res + ssilu(v)   3: v + res
// Requires K % 16 == 0, N % 64 == 0.  8 waves/block, 128x64 tile.
// Double-buffered LDS; A/B fragments are single aligned b64 loads.
// OOB A rows are clamped (their results are never stored), so the staging
// path has no predication / exec-mask traffic.
// ---------------------------------------------------------------------------
#define SA_STRIDE 18   // even => (k,k+1) pair 8B-aligned; breaks 16-float bank period

template<int EPI>
__global__ __launch_bounds__(256) void wmma_gemm_k(
    const float* __restrict__ A, const float* __restrict__ W,
    const float* __restrict__ bias, const float* __restrict__ res,
    float* __restrict__ out, int M, int K, int N)
{
  __shared__ float sA[2][128 * SA_STRIDE];
  __shared__ float sW[2][16 * 64];     // k-paired: [(k>>1)*64 + n]*2 + (k&1)
  __shared__ float sB[64];
  const int tid  = threadIdx.x;
  const int lane = tid & 31;
  const int wv   = tid >> 5;
  const int m0   = blockIdx.x * 128;
  const int n0   = blockIdx.y * 64;
  if (tid < 64) sB[tid] = bias ? bias[n0 + tid] : 0.0f;
  const int kb  = (lane & 16) ? 2 : 0;   // K base for this half-wave (frag layout)
  const int l15 = lane & 15;

  const float* Ab   = A + (size_t)m0 * K;   // 64-bit math once per block
  const int    mrem = M - m0;               // rows valid in this tile (>=1)
  const int    mcl  = mrem - 1;

  float ar[8];   // staged A values (registers)
  float wr[4];   // staged W values

  v8f acc[4] = {};

  auto LOADS = [&](int k0) {
#pragma unroll
    for (int i = 0; i < 8; ++i) {          // A tile 128x16, clamped rows, coalesced
      int idx = tid + i * 256;
      int r = idx >> 4, kk = idx & 15;
      int rc = r < mrem ? r : mcl;
      ar[i] = Ab[rc * K + (k0 + kk)];
    }
#pragma unroll
    for (int i = 0; i < 4; ++i) {          // W tile 16x64, coalesced
      int idx = tid + i * 256;
      int kk = idx >> 6, n = idx & 63;
      wr[i] = W[(size_t)(k0 + kk) * N + (n0 + n)];
    }
  };
  auto STORES = [&](int bf) {
#pragma unroll
    for (int i = 0; i < 8; ++i) {
      int idx = tid + i * 256;
      int r = idx >> 4, kk = idx & 15;
      sA[bf][r * SA_STRIDE + kk] = ar[i];
    }
#pragma unroll
    for (int i = 0; i < 4; ++i) {
      int idx = tid + i * 256;
      int kk = idx >> 6, n = idx & 63;
      sW[bf][((kk >> 1) * 64 + n) * 2 + (kk & 1)] = wr[i];
    }
  };
  auto COMPUTE = [&](int bf) {
#pragma unroll
    for (int kk = 0; kk < 16; kk += 4) {
      // A frag: lane m=l15, k = kk+kb, kk+kb+1  (contiguous, 8B aligned)
      v2f a = *(const v2f*)&sA[bf][(wv * 16 + l15) * SA_STRIDE + kk + kb];
#pragma unroll
      for (int nt = 0; nt < 4; ++nt) {
        // B frag: lane n, k pair (kk+kb, kk+kb+1) contiguous in paired layout
        v2f b = *(const v2f*)&sW[bf][(((kk + kb) >> 1) * 64 + nt * 16 + l15) * 2];
        acc[nt] = __builtin_amdgcn_wmma_f32_16x16x4_f32(
            false, a, false, b, (short)0, acc[nt], false, false);
      }
    }
  };

  LOADS(0);
  STORES(0);
  __syncthreads();
  const int nk = K >> 4;
  for (int t = 0; t < nk; ++t) {
    if (t + 1 < nk) LOADS((t + 1) * 16);   // prefetch next chunk under WMMAs
    COMPUTE(t & 1);
    if (t + 1 < nk) {
      STORES((t + 1) & 1);
      __syncthreads();
    }
  }

  const int mloc = wv * 16 + ((lane & 16) ? 8 : 0);   // row within tile
  float*       Ob = out + (size_t)m0 * N;
  const float* Rb = (EPI == 2 || EPI == 3) ? res + (size_t)m0 * N : nullptr;
#pragma unroll
  for (int nt = 0; nt < 4; ++nt) {
    const int   n  = n0 + nt * 16 + l15;
    const float bb = sB[nt * 16 + l15];
#pragma unroll
    for (int r = 0; r < 8; ++r) {
      int mr = mloc + r;
      if (mr < mrem) {
        float v = acc[nt][r] + bb;
        int oi = mr * N + n;                 // 32-bit in-tile offset
        if (EPI == 1)      v = ssilu_f(v);
        else if (EPI == 2) v = Rb[oi] + ssilu_f(v);
        else if (EPI == 3) v = v + Rb[oi];
        Ob[oi] = v;
      }
    }
  }
}

// ---------------------------------------------------------------------------
// Elementwise / scatter kernels
// ---------------------------------------------------------------------------
__global__ void k_zero(float* __restrict__ p, long long n) {
  long long i = (long long)blockIdx.x * blockDim.x + threadIdx.x;
  if (i < n) p[i] = 0.0f;
}

__global__ void k_init_nodes(const int* __restrict__ z, const float* __restrict__ emb,
                             float* __restrict__ x) {
  int n = blockIdx.x, c = threadIdx.x;
  x[(size_t)n * H + c] = emb[(size_t)z[n] * H + c];
}

__global__ void k_init_vx(const float* __restrict__ vn, float* __restrict__ vx) {
  vx[(size_t)blockIdx.x * H + threadIdx.x] = vn[threadIdx.x];
}

__global__ void k_count(const int* __restrict__ batch, float* __restrict__ cnt, int N) {
  int n = blockIdx.x * blockDim.x + threadIdx.x;
  if (n < N) atomicAdd(&cnt[batch[n]], 1.0f);
}

__global__ void k_concat_xg(const float* __restrict__ x, const float* __restrict__ vx,
                            const int* __restrict__ batch, float* __restrict__ t2) {
  int n = blockIdx.x, c = threadIdx.x;
  t2[(size_t)n * 256 + c]     = x[(size_t)n * H + c];
  t2[(size_t)n * 256 + H + c] = vx[(size_t)batch[n] * H + c];
}

__global__ void k_vec_plus_g(const float* __restrict__ vec, const float* __restrict__ vvec,
                             const int* __restrict__ batch, float* __restrict__ tv) {
  int n = blockIdx.x, c = threadIdx.x;
  int g = batch[n];
#pragma unroll
  for (int d = 0; d < 3; ++d)
    tv[((size_t)n * 3 + d) * H + c] =
        vec[((size_t)n * 3 + d) * H + c] + vvec[((size_t)g * 3 + d) * H + c];
}

__global__ void k_message(const int* __restrict__ jj, const int* __restrict__ ii,
                          const float* __restrict__ rbfh, const float* __restrict__ ev,
                          const float* __restrict__ xh, const float* __restrict__ vec,
                          float* __restrict__ dvec, float* __restrict__ dx) {
  int e = blockIdx.x, c = threadIdx.x;
  int j = jj[e], i = ii[e];
  const float* rh = rbfh + (size_t)e * H3;
  const float* xr = xh + (size_t)j * H3;
  const float INV3 = 0.57735026919f, INVH = 0.08838834765f;
  float x1 = xr[c]         * rh[c]         * INV3;
  float x2 = xr[H + c]     * rh[H + c]     * INV3;
  float x3 = xr[2 * H + c] * rh[2 * H + c] * INV3;
  atomicAdd(&dx[(size_t)i * H + c], x3);
#pragma unroll
  for (int d = 0; d < 3; ++d) {
    float e_d = ev[(size_t)e * 3 + d];
    float vj  = vec[((size_t)j * 3 + d) * H + c];
    atomicAdd(&dvec[((size_t)i * 3 + d) * H + c], (x1 * vj + x2 * e_d) * INVH);
  }
}

__global__ void k_node_fin(float* __restrict__ x, float* __restrict__ vec,
                           const float* __restrict__ dx, const float* __restrict__ dvec) {
  int n = blockIdx.x, c = threadIdx.x;
  x[(size_t)n * H + c] = (x[(size_t)n * H + c] + dx[(size_t)n * H + c]) * 0.70710678119f;
#pragma unroll
  for (int d = 0; d < 3; ++d) {
    size_t idx = ((size_t)n * 3 + d) * H + c;
    vec[idx] += dvec[idx];
  }
}

__global__ void k_norm_concat(const float* __restrict__ x, const float* __restrict__ vv12,
                              float* __restrict__ t2) {
  int n = blockIdx.x, c = threadIdx.x;
  float s = 0.0f;
#pragma unroll
  for (int d = 0; d < 3; ++d) {
    float v = vv12[((size_t)n * 3 + d) * 256 + H + c];  // vec2 half
    s += v * v;
  }
  t2[(size_t)n * 256 + c]     = x[(size_t)n * H + c];
  t2[(size_t)n * 256 + H + c] = sqrtf(s + 1e-8f);
}

__global__ void k_update_apply(float* __restrict__ x, float* __restrict__ vec,
                               const float* __restrict__ h3, const float* __restrict__ vv12) {
  int n = blockIdx.x, c = threadIdx.x;
  float xv1 = h3[(size_t)n * H3 + c];
  float xv2 = h3[(size_t)n * H3 + H + c];
  float xv3 = h3[(size_t)n * H3 + 2 * H + c];
  float xo  = x[(size_t)n * H + c];
  x[(size_t)n * H + c] = xo + xv2 * 0.70710678119f + xo * tanhf(xv3);
#pragma unroll
  for (int d = 0; d < 3; ++d)
    vec[((size_t)n * 3 + d) * H + c] += xv1 * vv12[((size_t)n * 3 + d) * 256 + c];  // vec1 half
}

__global__ void k_pool(const float* __restrict__ x, const float* __restrict__ vec,
                       const int* __restrict__ batch, float* __restrict__ px,
                       float* __restrict__ pvec) {
  int n = blockIdx.x, c = threadIdx.x;
  int g = batch[n];
  atomicAdd(&px[(size_t)g * H + c], x[(size_t)n * H + c]);
#pragma unroll
  for (int d = 0; d < 3; ++d)
    atomicAdd(&pvec[((size_t)g * 3 + d) * H + c], vec[((size_t)n * 3 + d) * H + c]);
}

__global__ void k_global_pre(const float* __restrict__ px, const float* __restrict__ pvec,
                             const float* __restrict__ vx, const float* __restrict__ vvec,
                             const float* __restrict__ cnt, float* __restrict__ gh,
                             float* __restrict__ ptmp) {
  int g = blockIdx.x, c = threadIdx.x;
  float inv = 1.0f / fmaxf(cnt[g], 1.0f);
  gh[(size_t)g * 256 + c]     = px[(size_t)g * H + c] * inv;
  gh[(size_t)g * 256 + H + c] = vx[(size_t)g * H + c];
#pragma unroll
  for (int d = 0; d < 3; ++d) {
    size_t idx = ((size_t)g * 3 + d) * H + c;
    ptmp[idx] = pvec[idx] * inv + vvec[idx];
  }
}

__global__ void k_energy(const float* __restrict__ t1, const float* __restrict__ w2,
                         const float* __restrict__ b2, const int* __restrict__ batch,
                         float* __restrict__ out, int N) {
  int n = blockIdx.x * blockDim.x + threadIdx.x;
  if (n >= N) return;
  float s = b2[0];
  const float* row = t1 + (size_t)n * H2;
#pragma unroll
  for (int k = 0; k < H2; ++k) s += row[k] * w2[k];
  atomicAdd(&out[batch[n]], s);
}

// ---------------------------------------------------------------------------
// Host side
// ---------------------------------------------------------------------------
static inline void gemm(hipStream_t s, int epi, const float* A, const float* W,
                        const float* b, const float* res, float* out, int M, int K, int N) {
  dim3 grid((unsigned)((M + 127) / 128), (unsigned)(N / 64)), blk(256);
  switch (epi) {
    case 0: wmma_gemm_k<0><<<grid, blk, 0, s>>>(A, W, b, res, out, M, K, N); break;
    case 1: wmma_gemm_k<1><<<grid, blk, 0, s>>>(A, W, b, res, out, M, K, N); break;
    case 2: wmma_gemm_k<2><<<grid, blk, 0, s>>>(A, W, b, res, out, M, K, N); break;
    case 3: wmma_gemm_k<3><<<grid, blk, 0, s>>>(A, W, b, res, out, M, K, N); break;
  }
}
static inline void zero_f(hipStream_t s, float* p, long long n) {
  k_zero<<<(unsigned)((n + 255) / 256), 256, 0, s>>>(p, n);
}

extern "C" void kernel_launch(void* const* d_in, const int* in_sizes, int n_in,
                              void* d_out, int out_size, void* d_ws, size_t ws_size,
                              hipStream_t stream) {
  const int*   z     = (const int*)d_in[0];
  const int*   batch = (const int*)d_in[1];
  const int*   eidx  = (const int*)d_in[2];   // [2,E] flat: j = eidx[e], i = eidx[E+e]
  const float* erbf  = (const float*)d_in[3]; // [E,R]
  const float* evec  = (const float*)d_in[4]; // [E,3]
  // d_in[5] = num_graphs (device scalar, B taken from out_size)
  const float* atom_emb = (const float*)d_in[6];
  const float* vn_emb   = (const float*)d_in[7];
  const float* msg_w1   = (const float*)d_in[8];
  const float* msg_b1   = (const float*)d_in[9];
  const float* msg_w2   = (const float*)d_in[10];
  const float* msg_b2   = (const float*)d_in[11];
  const float* rbf_w    = (const float*)d_in[12];
  const float* rbf_b    = (const float*)d_in[13];
  const float* upd_vec_w= (const float*)d_in[14];
  const float* upd_w1   = (const float*)d_in[15];
  const float* upd_b1   = (const float*)d_in[16];
  const float* upd_w2   = (const float*)d_in[17];
  const float* upd_b2   = (const float*)d_in[18];
  const float* gs_vn_w1 = (const float*)d_in[19];
  const float* gs_vn_b1 = (const float*)d_in[20];
  const float* gs_vn_w2 = (const float*)d_in[21];
  const float* gs_vn_b2 = (const float*)d_in[22];
  const float* gs_nd_w1 = (const float*)d_in[23];
  const float* gs_nd_b1 = (const float*)d_in[24];
  const float* gs_nd_w2 = (const float*)d_in[25];
  const float* gs_nd_b2 = (const float*)d_in[26];
  const float* gv_vn_w  = (const float*)d_in[27];
  const float* gv_nd_w  = (const float*)d_in[28];
  const float* out_w1   = (const float*)d_in[29];
  const float* out_b1   = (const float*)d_in[30];
  const float* out_w2   = (const float*)d_in[31];
  const float* out_b2   = (const float*)d_in[32];

  const int N = in_sizes[0];
  const int E = in_sizes[2] / 2;
  const int B = out_size;

  // ---- workspace layout (floats) ----
  float* w = (float*)d_ws;
  size_t o = 0;
  float* x    = w + o; o += (size_t)N * H;         // node scalar features
  float* vec  = w + o; o += (size_t)N * H3;        // node vector features [N,3,H]
  float* t1   = w + o; o += (size_t)N * H;         // MLP hidden / out1
  float* t2   = w + o; o += (size_t)N * 2 * H;     // concat buffers [N,2H]
  float* xh   = w + o; o += (size_t)N * H3;        // x_proj output / h3 / tv
  float* vv12 = w + o; o += (size_t)N * 3 * 2 * H; // [N,3,2H] vec1|vec2
  const int EC = 40000;                            // edge chunk
  float* rbfh = w + o; o += (size_t)EC * H3;       // rbf_proj chunk
  float* px   = w + o; o += (size_t)B * H;
  float* pvec = w + o; o += (size_t)B * H3;
  float* vx   = w + o; o += (size_t)B * H;
  float* vvec = w + o; o += (size_t)B * H3;
  float* gh   = w + o; o += (size_t)B * 2 * H;
  float* gt   = w + o; o += (size_t)B * H;
  float* ptmp = w + o; o += (size_t)B * H3;
  float* cnt  = w + o; o += (size_t)B;
  float* dvec = vv12;   // alias: dead before vv12 is written
  float* dx   = t1;     // alias: dead before t1 reused

  // ---- init ----
  k_init_nodes<<<N, 128, 0, stream>>>(z, atom_emb, x);
  zero_f(stream, vec, (long long)N * H3);
  k_init_vx<<<B, 128, 0, stream>>>(vn_emb, vx);
  zero_f(stream, vvec, (long long)B * H3);
  zero_f(stream, cnt, B);
  k_count<<<(N + 255) / 256, 256, 0, stream>>>(batch, cnt, N);

  for (int l = 0; l < 3; ++l) {
    const float* mw1 = msg_w1 + (size_t)l * H * H2;
    const float* mb1 = msg_b1 + (size_t)l * H2;
    const float* mw2 = msg_w2 + (size_t)l * H2 * H3;
    const float* mb2 = msg_b2 + (size_t)l * H3;
    const float* rw  = rbf_w  + (size_t)l * RDIM * H3;
    const float* rb  = rbf_b  + (size_t)l * H3;
    const float* uvw = upd_vec_w + (size_t)l * H * 2 * H;
    const float* uw1 = upd_w1 + (size_t)l * 2 * H * H;
    const float* ub1 = upd_b1 + (size_t)l * H;
    const float* uw2 = upd_w2 + (size_t)l * H * H3;
    const float* ub2 = upd_b2 + (size_t)l * H3;
    const float* svw1 = gs_vn_w1 + (size_t)l * 2 * H * H;
    const float* svb1 = gs_vn_b1 + (size_t)l * H;
    const float* svw2 = gs_vn_w2 + (size_t)l * H * H;
    const float* svb2 = gs_vn_b2 + (size_t)l * H;
    const float* snw1 = gs_nd_w1 + (size_t)l * 2 * H * H;
    const float* snb1 = gs_nd_b1 + (size_t)l * H;
    const float* snw2 = gs_nd_w2 + (size_t)l * H * H;
    const float* snb2 = gs_nd_b2 + (size_t)l * H;
    const float* gvw  = gv_vn_w + (size_t)l * H * H;
    const float* gnw  = gv_nd_w + (size_t)l * H * H;

    // GlobalScalar.update_local_emb:  x += ssilu(ssilu([x|vx[b]]@w1+b1)@w2+b2)
    k_concat_xg<<<N, 128, 0, stream>>>(x, vx, batch, t2);
    gemm(stream, 1, t2, snw1, snb1, nullptr, t1, N, 2 * H, H);
    gemm(stream, 2, t1, snw2, snb2, x, x, N, H, H);

    // GlobalVector.update_local_emb:  vec = (vec+vvec[b])@w + vec
    k_vec_plus_g<<<N, 128, 0, stream>>>(vec, vvec, batch, xh);
    gemm(stream, 3, xh, gnw, nullptr, vec, vec, 3 * N, H, H);

    // Message projections
    gemm(stream, 1, x, mw1, mb1, nullptr, t1, N, H, H2);
    gemm(stream, 0, t1, mw2, mb2, nullptr, xh, N, H2, H3);

    // Edge scatter (chunked rbf_proj GEMM + gather/scatter)
    zero_f(stream, dx, (long long)N * H);
    zero_f(stream, dvec, (long long)N * H3);
    for (int e0 = 0; e0 < E; e0 += EC) {
      int ec = (E - e0 < EC) ? (E - e0) : EC;
      gemm(stream, 0, erbf + (size_t)e0 * RDIM, rw, rb, nullptr, rbfh, ec, RDIM, H3);
      k_message<<<ec, 128, 0, stream>>>(eidx + e0, eidx + E + e0, rbfh,
                                        evec + (size_t)e0 * 3, xh, vec, dvec, dx);
    }
    k_node_fin<<<N, 128, 0, stream>>>(x, vec, dx, dvec);

    // E2GNNUpdate
    gemm(stream, 0, vec, uvw, nullptr, nullptr, vv12, 3 * N, H, 2 * H);
    k_norm_concat<<<N, 128, 0, stream>>>(x, vv12, t2);
    gemm(stream, 1, t2, uw1, ub1, nullptr, t1, N, 2 * H, H);
    gemm(stream, 0, t1, uw2, ub2, nullptr, xh, N, H, H3);
    k_update_apply<<<N, 128, 0, stream>>>(x, vec, xh, vv12);

    // Global updates
    zero_f(stream, px, (long long)B * H);
    zero_f(stream, pvec, (long long)B * H3);
    k_pool<<<N, 128, 0, stream>>>(x, vec, batch, px, pvec);
    k_global_pre<<<B, 128, 0, stream>>>(px, pvec, vx, vvec, cnt, gh, ptmp);
    gemm(stream, 1, gh, svw1, svb1, nullptr, gt, B, 2 * H, H);
    gemm(stream, 2, gt, svw2, svb2, vx, vx, B, H, H);
    gemm(stream, 3, ptmp, gvw, nullptr, vvec, vvec, 3 * B, H, H);
  }

  // Output head: per-atom energy -> segment sum over graphs
  gemm(stream, 1, x, out_w1, out_b1, nullptr, t1, N, H, H2);
  zero_f(stream, (float*)d_out, B);
  k_energy<<<(N + 127) / 128, 128, 0, stream>>>(t1, out_w2, out_b2, batch,
                                                (float*)d_out, N);
  (void)n_in; (void)ws_size;
}